// XConv_42262478193074
// MI455X (gfx1250) — compile-verified
//
#include <hip/hip_runtime.h>
#include <hip/hip_bf16.h>
#include <math.h>

// Problem constants (match reference)
#define N_PTS 16384
#define C_IN  64
#define C_OUT 128
#define KNN   16
#define DM    2
#define KD    48    // KNN * D
#define KK    256   // KNN * KNN

typedef __attribute__((ext_vector_type(2))) float v2f;
typedef __attribute__((ext_vector_type(8))) float v8f;

// D(16x16,f32) = A(16x4,f32) * B(4x16,f32) + C  -- full fp32 precision matrix op.
// Layouts (ISA 7.12.2): A: lane L(0..15) holds M=L, {K0,K1}; lanes 16..31 hold {K2,K3}.
// B: lane L holds N=L&15, rows {K0,K1} (half 0) / {K2,K3} (half 1).
// C/D: vgpr r -> M = r + 8*(lane>>4), N = lane&15.
__device__ __forceinline__ v8f wmma_f32_k4(v2f a, v2f b, v8f c) {
  return __builtin_amdgcn_wmma_f32_16x16x4_f32(
      /*neg_a=*/false, a, /*neg_b=*/false, b,
      /*c_mod=*/(short)0, c, /*reuse_a=*/false, /*reuse_b=*/false);
}

__device__ __forceinline__ float elu1(float v) {
  return v > 0.0f ? v : (expf(v) - 1.0f);
}

// ---------------------------------------------------------------------------
// Kernel 1: exact KNN (K=16) without materializing the 1 GB NxN matrix.
// One thread per query point; candidate tiles staged in LDS; top-16 kept
// sorted in registers via a fully unrolled compare-swap insertion chain
// (static indices only -> no scratch). Candidates scanned in index order with
// strict '<', reproducing top_k's lowest-index tie preference.
// ---------------------------------------------------------------------------
#define KTILE 256
__global__ __launch_bounds__(256) void knn_kernel(const float* __restrict__ pos,
                                                  int* __restrict__ col) {
  __shared__ float sx[KTILE], sy[KTILE], sz[KTILE];
  const int q = blockIdx.x * 256 + threadIdx.x;
  const float qx = pos[q * 3 + 0], qy = pos[q * 3 + 1], qz = pos[q * 3 + 2];

  float bd[KNN];
  int   bi[KNN];
#pragma unroll
  for (int i = 0; i < KNN; ++i) { bd[i] = 3.4028235e38f; bi[i] = 0; }

  for (int base = 0; base < N_PTS; base += KTILE) {
    __syncthreads();
    const int t = threadIdx.x;
    sx[t] = pos[(base + t) * 3 + 0];
    sy[t] = pos[(base + t) * 3 + 1];
    sz[t] = pos[(base + t) * 3 + 2];
    if (base + KTILE < N_PTS)
      __builtin_prefetch(pos + (size_t)(base + KTILE) * 3, 0, 0);
    __syncthreads();
#pragma unroll 4
    for (int j = 0; j < KTILE; ++j) {
      const float dx = sx[j] - qx, dy = sy[j] - qy, dz = sz[j] - qz;
      const float d = dx * dx + dy * dy + dz * dz;
      if (d < bd[KNN - 1]) {          // beats current worst -> insert sorted
        float cd = d; int ci = base + j;
#pragma unroll
        for (int i = KNN - 1; i >= 0; --i) {
          if (cd < bd[i]) {
            const float td = bd[i]; bd[i] = cd; cd = td;
            const int   ti = bi[i]; bi[i] = ci; ci = ti;
          }
        }
      }
    }
  }
#pragma unroll
  for (int i = 0; i < KNN; ++i) col[q * KNN + i] = bi[i];
}

// ---------------------------------------------------------------------------
// Kernel 2: per-point transform generator. One wave == 16 points.
//   h = elu(rel[16x48] @ W1[48x256] + b1)          (16 tiles x 12 K-steps WMMA)
//   h = elu(grouped 16x16 conv1)                    (16 groups x 4 K-steps)
//   T =      grouped 16x16 conv2                    (16 groups x 4 K-steps)
// 2 waves per block; weights stay in L2 (tiny), activations in LDS.
// ---------------------------------------------------------------------------
__global__ __launch_bounds__(64) void mlp_kernel(
    const float* __restrict__ pos, const int* __restrict__ col,
    const float* __restrict__ W1,  const float* __restrict__ b1,
    const float* __restrict__ c1w, const float* __restrict__ c1b,
    const float* __restrict__ c2w, const float* __restrict__ c2b,
    float* __restrict__ Tout) {
  __shared__ float relS[2][16][KD];   // 6 KB
  __shared__ float hS[2][16][KK];     // 32 KB
  const int wave = threadIdx.x >> 5;
  const int lane = threadIdx.x & 31;
  const int row  = lane & 15;
  const int half = lane >> 4;
  const int p0   = blockIdx.x * 32 + wave * 16;

  // Stage relative positions: rel[pt][nb*3+c] = pos[col[pt][nb]] - pos[pt]
  for (int idx = lane; idx < 256; idx += 32) {
    const int pt = idx >> 4, nb = idx & 15;
    const int c  = col[(p0 + pt) * KNN + nb];
    relS[wave][pt][nb * 3 + 0] = pos[c * 3 + 0] - pos[(p0 + pt) * 3 + 0];
    relS[wave][pt][nb * 3 + 1] = pos[c * 3 + 1] - pos[(p0 + pt) * 3 + 1];
    relS[wave][pt][nb * 3 + 2] = pos[c * 3 + 2] - pos[(p0 + pt) * 3 + 2];
  }
  __syncthreads();

  // Layer 1: [16 x 48] @ [48 x 256]
  for (int nt = 0; nt < 16; ++nt) {
    const int ncol = nt * 16 + row;
    v8f acc = {};
#pragma unroll
    for (int kk = 0; kk < 12; ++kk) {
      const int kb = kk * 4 + 2 * half;
      v2f a = { relS[wave][row][kb], relS[wave][row][kb + 1] };
      v2f b = { W1[kb * KK + ncol], W1[(kb + 1) * KK + ncol] };
      acc = wmma_f32_k4(a, b, acc);
    }
#pragma unroll
    for (int r = 0; r < 8; ++r)
      hS[wave][r + 8 * half][ncol] = elu1(acc[r] + b1[ncol]);
  }
  __syncthreads();

  // Grouped conv1: out[pt][g][j] = elu(sum_t h[pt][g][t] * c1w[g][j][t] + c1b[g][j])
  // In-place per group is safe: within a wave DS ops execute in order, and all
  // A-tile loads for group g precede the group-g stores.
  for (int g = 0; g < 16; ++g) {
    v8f acc = {};
#pragma unroll
    for (int kk = 0; kk < 4; ++kk) {
      const int kb = kk * 4 + 2 * half;
      v2f a = { hS[wave][row][g * 16 + kb], hS[wave][row][g * 16 + kb + 1] };
      v2f b = { c1w[g * 256 + row * 16 + kb], c1w[g * 256 + row * 16 + kb + 1] };
      acc = wmma_f32_k4(a, b, acc);
    }
#pragma unroll
    for (int r = 0; r < 8; ++r)
      hS[wave][r + 8 * half][g * 16 + row] = elu1(acc[r] + c1b[g * 16 + row]);
  }
  __syncthreads();

  // Grouped conv2 -> T (no activation), written straight to global workspace.
  for (int g = 0; g < 16; ++g) {
    v8f acc = {};
#pragma unroll
    for (int kk = 0; kk < 4; ++kk) {
      const int kb = kk * 4 + 2 * half;
      v2f a = { hS[wave][row][g * 16 + kb], hS[wave][row][g * 16 + kb + 1] };
      v2f b = { c2w[g * 256 + row * 16 + kb], c2w[g * 256 + row * 16 + kb + 1] };
      acc = wmma_f32_k4(a, b, acc);
    }
#pragma unroll
    for (int r = 0; r < 8; ++r)
      Tout[(size_t)(p0 + r + 8 * half) * KK + g * 16 + row] =
          acc[r] + c2b[g * 16 + row];
  }
}

// ---------------------------------------------------------------------------
// Kernel 3: xt = x_star @ T^T per point (4 M-tiles x 4 K-steps WMMA), then the
// grouped DM-conv reduction over k -> Y[N,128]. One wave per point, 8/block.
// LDS row stride 17 breaks the 4-bank conflict pattern of stride 16.
// ---------------------------------------------------------------------------
__global__ __launch_bounds__(256) void xform_kernel(
    const float* __restrict__ x, const int* __restrict__ col,
    const float* __restrict__ Tin,
    const float* __restrict__ conv_w, const float* __restrict__ conv_b,
    float* __restrict__ Y) {
  __shared__ float Tl[8][KK];          // 8 KB
  __shared__ float xs[8][C_IN * 17];   // ~34 KB
  const int wave = threadIdx.x >> 5;
  const int lane = threadIdx.x & 31;
  const int row  = lane & 15;
  const int half = lane >> 4;
  const int p    = blockIdx.x * 8 + wave;

#pragma unroll
  for (int i = 0; i < 8; ++i)
    Tl[wave][lane + 32 * i] = Tin[(size_t)p * KK + lane + 32 * i];

  // Gather x_star[ch][j] = x[col[p][j]][ch] (coalesced over ch)
  for (int idx = lane; idx < C_IN * KNN; idx += 32) {
    const int ch = idx & 63, j = idx >> 6;
    const int cj = col[p * KNN + j];
    xs[wave][ch * 17 + j] = x[(size_t)cj * C_IN + ch];
  }
  __syncthreads();

  // xt[c][k'] = sum_j x_star[c][j] * T[k'][j]   (B[j][k'] = T[k'][j])
#pragma unroll
  for (int mt = 0; mt < 4; ++mt) {
    v8f acc = {};
#pragma unroll
    for (int kk = 0; kk < 4; ++kk) {
      const int kb = kk * 4 + 2 * half;
      v2f a = { xs[wave][(mt * 16 + row) * 17 + kb],
                xs[wave][(mt * 16 + row) * 17 + kb + 1] };
      v2f b = { Tl[wave][row * 16 + kb], Tl[wave][row * 16 + kb + 1] };
      acc = wmma_f32_k4(a, b, acc);
    }
#pragma unroll
    for (int r = 0; r < 8; ++r)
      xs[wave][(mt * 16 + r + 8 * half) * 17 + row] = acc[r];  // in place
  }
  __syncthreads();

  // y[c*2+d] = sum_k xt[c][k] * conv_w[c][d][k] + conv_b[c][d]
#pragma unroll
  for (int t = 0; t < 4; ++t) {
    const int o = lane + 32 * t;            // 0..127
    const int ch = o >> 1, d = o & 1;
    float acc = conv_b[ch * DM + d];
#pragma unroll
    for (int k = 0; k < KNN; ++k)
      acc += xs[wave][ch * 17 + k] * conv_w[ch * (DM * KNN) + d * KNN + k];
    Y[(size_t)p * C_OUT + o] = acc;
  }
}

// ---------------------------------------------------------------------------
// Kernel 4: out[N,128] = Y[N,128] @ lin_w[128,128] + lin_b.
// One 16x16 output tile per wave, 32 f32 K-steps. Y + lin_w are L2-resident.
// ---------------------------------------------------------------------------
__global__ __launch_bounds__(256) void out_gemm_kernel(
    const float* __restrict__ Y, const float* __restrict__ lin_w,
    const float* __restrict__ lin_b, float* __restrict__ out) {
  const int wave = threadIdx.x >> 5;
  const int lane = threadIdx.x & 31;
  const int row  = lane & 15;
  const int half = lane >> 4;
  const int tile = blockIdx.x * 8 + wave;   // (N/16)*8 tiles total
  const int m0   = (tile >> 3) * 16;
  const int n0   = (tile & 7) * 16;
  const int ncol = n0 + row;

  v8f acc = {};
#pragma unroll
  for (int kk = 0; kk < 32; ++kk) {
    const int kb = kk * 4 + 2 * half;
    v2f a = { Y[(size_t)(m0 + row) * C_OUT + kb],
              Y[(size_t)(m0 + row) * C_OUT + kb + 1] };
    v2f b = { lin_w[kb * C_OUT + ncol], lin_w[(kb + 1) * C_OUT + ncol] };
    acc = wmma_f32_k4(a, b, acc);
  }
#pragma unroll
  for (int r = 0; r < 8; ++r)
    out[(size_t)(m0 + r + 8 * half) * C_OUT + ncol] = acc[r] + lin_b[ncol];
}

// ---------------------------------------------------------------------------
extern "C" void kernel_launch(void* const* d_in, const int* in_sizes, int n_in,
                              void* d_out, int out_size, void* d_ws, size_t ws_size,
                              hipStream_t stream) {
  (void)in_sizes; (void)n_in; (void)out_size; (void)ws_size;
  const float* x   = (const float*)d_in[0];
  const float* pos = (const float*)d_in[1];
  const float* W1  = (const float*)d_in[2];
  const float* b1  = (const float*)d_in[3];
  const float* c1w = (const float*)d_in[4];
  const float* c1b = (const float*)d_in[5];
  const float* c2w = (const float*)d_in[6];
  const float* c2b = (const float*)d_in[7];
  const float* cw  = (const float*)d_in[8];
  const float* cb  = (const float*)d_in[9];
  const float* lw  = (const float*)d_in[10];
  const float* lb  = (const float*)d_in[11];
  float* out = (float*)d_out;

  // Workspace layout: col[N*16] int (1 MB) | T[N*256] f32 (16 MB) | Y[N*128] f32 (8 MB)
  int*   col = (int*)d_ws;
  float* T   = (float*)((char*)d_ws + (size_t)N_PTS * KNN * sizeof(int));
  float* Y   = T + (size_t)N_PTS * KK;

  knn_kernel<<<N_PTS / 256, 256, 0, stream>>>(pos, col);
  mlp_kernel<<<N_PTS / 32, 64, 0, stream>>>(pos, col, W1, b1, c1w, c1b, c2w, c2b, T);
  xform_kernel<<<N_PTS / 8, 256, 0, stream>>>(x, col, T, cw, cb, Y);
  out_gemm_kernel<<<(N_PTS / 16) * 8 / 8, 256, 0, stream>>>(Y, lw, lb, out);
}